// VirtualNodeDgl_5205500363155
// MI455X (gfx1250) — compile-verified
//
#include <hip/hip_runtime.h>

#define D 128
#define MGRAPHS 4096

typedef __attribute__((ext_vector_type(2))) float v2f;
typedef __attribute__((ext_vector_type(8))) float v8f;

// ---------------------------------------------------------------------------
// Kernel 1: segment sum. seg is sorted, so block m binary-searches its row
// range [lo, hi) and accumulates without atomics (deterministic).
// blockDim = 128 (one column per thread -> coalesced 512B row loads).
// ---------------------------------------------------------------------------
__global__ void vn_segsum_kernel(const float* __restrict__ h,
                                 const int* __restrict__ seg,
                                 float* __restrict__ pool,
                                 int n_nodes) {
    const int m = blockIdx.x;           // graph id, 0..M-1
    const int t = threadIdx.x;          // column, 0..127

    // lower_bound(seg, m) and lower_bound(seg, m+1); uniform across block,
    // served by scalar/constant caches.
    int lo = 0, hi = n_nodes;
    while (lo < hi) { int mid = (lo + hi) >> 1; if (seg[mid] < m) lo = mid + 1; else hi = mid; }
    int lo2 = lo, hi2 = n_nodes;
    while (lo2 < hi2) { int mid = (lo2 + hi2) >> 1; if (seg[mid] < m + 1) lo2 = mid + 1; else hi2 = mid; }
    const int row_end = lo2;

    float a0 = 0.f, a1 = 0.f, a2 = 0.f, a3 = 0.f;
    long long r = lo;
    // 4x unroll: 4 independent loads in flight per thread.
    for (; r + 4 <= row_end; r += 4) {
        float x0 = h[(r + 0) * D + t];
        float x1 = h[(r + 1) * D + t];
        float x2 = h[(r + 2) * D + t];
        float x3 = h[(r + 3) * D + t];
        a0 += x0; a1 += x1; a2 += x2; a3 += x3;
    }
    for (; r < row_end; ++r) a0 += h[r * D + t];

    pool[(long long)m * D + t] = (a0 + a1) + (a2 + a3);
}

// ---------------------------------------------------------------------------
// Kernel 2: vn_new = vn_h + relu((vn_h + pool) @ W^T + b), via
// V_WMMA_F32_16X16X4_F32. One 16x16 output tile per wave; K=128 -> 32 WMMAs.
// grid 256 blocks x 256 threads = 2048 waves = (4096/16) * (128/16) tiles.
// ---------------------------------------------------------------------------
__global__ void vn_gemm_wmma_kernel(const float* __restrict__ vn_h,
                                    const float* __restrict__ pool,
                                    const float* __restrict__ W,
                                    const float* __restrict__ bias,
                                    float* __restrict__ vn_out) {
    const int wave = (int)((blockIdx.x * blockDim.x + threadIdx.x) >> 5);
    const int lane = (int)(threadIdx.x & 31);

    const int tm = (wave >> 3) * 16;      // 256 tiles along M=4096
    const int tn = (wave & 7) * 16;       // 8 tiles along N=128

    const int lrow  = lane & 15;          // row-in-tile for A, col-in-tile for B
    const int khalf = (lane >> 4) * 2;    // lanes 16..31 carry K+2,K+3

    // A: X[tm+lrow][k0+khalf .. +1] with X = vn_h + pool (computed on the fly)
    const float* xa_vn = vn_h + (long long)(tm + lrow) * D;
    const float* xa_pl = pool + (long long)(tm + lrow) * D;
    // B: B[k][n] = W[n][k]  ->  lane loads W[tn+lrow][k0+khalf .. +1]
    const float* wb = W + (long long)(tn + lrow) * D;

    v8f c = {};
    #pragma unroll
    for (int k0 = 0; k0 < D; k0 += 4) {
        const int k = k0 + khalf;
        v2f a, b;
        a.x = xa_vn[k]     + xa_pl[k];
        a.y = xa_vn[k + 1] + xa_pl[k + 1];
        b.x = wb[k];
        b.y = wb[k + 1];
        c = __builtin_amdgcn_wmma_f32_16x16x4_f32(
                /*neg_a=*/false, a, /*neg_b=*/false, b,
                /*c_mod=*/(short)0, c, /*reuse_a=*/false, /*reuse_b=*/false);
    }

    // C/D layout: VGPR v -> M = tm + v (+8 for lanes 16..31), N = tn + (lane&15)
    const int n     = tn + lrow;
    const int mbase = tm + ((lane >> 4) << 3);
    const float bn  = bias[n];
    #pragma unroll
    for (int v = 0; v < 8; ++v) {
        const int m = mbase + v;
        float t = c[v] + bn;
        t = t > 0.f ? t : 0.f;                       // relu
        vn_out[(long long)m * D + n] = vn_h[(long long)m * D + n] + t;  // residual
    }
}

// ---------------------------------------------------------------------------
// Kernel 3: h_new = h + vn_new[seg].  float4 grid-stride; vn_new (2 MB) stays
// resident in L2 so the gather is nearly free; h streamed at full HBM rate.
// ---------------------------------------------------------------------------
__global__ void vn_bcast_add_kernel(const float* __restrict__ h,
                                    const int* __restrict__ seg,
                                    const float* __restrict__ vn_new,
                                    float* __restrict__ h_new,
                                    int n_nodes) {
    const long long total4 = (long long)n_nodes * (D / 4);   // float4 units
    const long long stride = (long long)gridDim.x * blockDim.x;
    for (long long i = (long long)blockIdx.x * blockDim.x + threadIdx.x;
         i < total4; i += stride) {
        const long long row = i >> 5;            // D/4 = 32 float4 per row
        const int c4 = (int)(i & 31);
        const int m = seg[row];
        const float4 hv = ((const float4*)h)[i];
        const float4 vv = *(const float4*)(vn_new + (long long)m * D + c4 * 4);
        float4 o;
        o.x = hv.x + vv.x; o.y = hv.y + vv.y;
        o.z = hv.z + vv.z; o.w = hv.w + vv.w;
        ((float4*)h_new)[i] = o;
    }
}

// ---------------------------------------------------------------------------
extern "C" void kernel_launch(void* const* d_in, const int* in_sizes, int n_in,
                              void* d_out, int out_size, void* d_ws, size_t ws_size,
                              hipStream_t stream) {
    const float* h    = (const float*)d_in[0];   // [N, 128]
    const float* vn_h = (const float*)d_in[1];   // [M, 128]
    const int*   seg  = (const int*)  d_in[2];   // [N], sorted
    const float* W    = (const float*)d_in[3];   // [128, 128]
    const float* bias = (const float*)d_in[4];   // [128]

    const int n_nodes = in_sizes[2];             // N = 1,000,000

    float* pool   = (float*)d_ws;                          // [M, 128] scratch
    float* h_new  = (float*)d_out;                         // [N, 128]
    float* vn_new = (float*)d_out + (long long)n_nodes * D;// [M, 128]

    // 1) per-graph segment sum (sorted seg -> binary search, no atomics)
    vn_segsum_kernel<<<MGRAPHS, 128, 0, stream>>>(h, seg, pool, n_nodes);

    // 2) virtual-node FC + relu + residual via fp32 WMMA
    //    (MGRAPHS/16)*(D/16) = 2048 tiles = 256 blocks * 8 waves
    vn_gemm_wmma_kernel<<<256, 256, 0, stream>>>(vn_h, pool, W, bias, vn_new);

    // 3) broadcast virtual node back to nodes
    vn_bcast_add_kernel<<<4096, 256, 0, stream>>>(h, seg, vn_new, h_new, n_nodes);
}